// GNN_26491358281754
// MI455X (gfx1250) — compile-verified
//
#include <hip/hip_runtime.h>

typedef __attribute__((ext_vector_type(2))) float v2f;
typedef __attribute__((ext_vector_type(8))) float v8f;

#define FEAT 64

// ---------------- degree / normalization ----------------
__global__ void k_deg_init(float* __restrict__ deg, int n) {
    int i = blockIdx.x * blockDim.x + threadIdx.x;
    if (i < n) deg[i] = 1.0f;  // self-loop contributes 1
}

__global__ void k_deg_edges(const long long* __restrict__ dst, float* __restrict__ deg, int E) {
    int e = blockIdx.x * blockDim.x + threadIdx.x;
    if (e < E) atomicAdd(&deg[(int)dst[e]], 1.0f);
}

__global__ void k_rsqrt_inplace(float* __restrict__ deg, int n) {
    int i = blockIdx.x * blockDim.x + threadIdx.x;
    if (i < n) deg[i] = rsqrtf(deg[i]);  // deg >= 1 always (self-loops)
}

// ---------------- fused dense transform + aggregation init ----------------
// xw   = relu?(A[M x K]) @ W[K x 64]                       -> out
// init = xw * dis^2 + b   (self-loop term + bias epilogue) -> initout
// One wave computes 16x16 f32 tiles via V_WMMA_F32_16X16X4_F32.
// Block = 128 threads = 4 waves; wave w owns output cols [16w,16w+16).
// B fragments (weights) preloaded into registers, reused across RTILES row tiles.
template <int K, bool RELU, int RTILES>
__global__ void k_gemm_wmma_fused(const float* __restrict__ A, const float* __restrict__ W,
                                  const float* __restrict__ bias, const float* __restrict__ dis,
                                  float* __restrict__ out, float* __restrict__ initout, int M) {
    const int lane = threadIdx.x & 31;
    const int wave = threadIdx.x >> 5;
    const int c0 = wave * 16;
    const int m  = lane & 15;          // row (A) / col (B) within tile
    const int kk = (lane >> 4) * 2;    // lane-half selects K pair {0,1} or {2,3}
    constexpr int KF = K / 4;

    const float bcol = bias[c0 + m];   // per-lane output-column bias

    // Preload this wave's B tile: 4x16 fragment per k-step (ISA 32-bit A/B layout)
    v2f bfrag[KF];
#pragma unroll
    for (int kf = 0; kf < KF; ++kf) {
        bfrag[kf].x = W[(size_t)(kf * 4 + kk    ) * FEAT + c0 + m];
        bfrag[kf].y = W[(size_t)(kf * 4 + kk + 1) * FEAT + c0 + m];
    }

    const int rblock = blockIdx.x * (16 * RTILES);
#pragma unroll
    for (int rt = 0; rt < RTILES; ++rt) {
        const int r0 = rblock + rt * 16;
        if (r0 >= M) break;
        const int arow = r0 + m;
        const int arow_c = arow < M ? arow : (M - 1);   // clamp loads; stores guarded
        const float* ap = A + (size_t)arow_c * K;

        v8f acc = {};
#pragma unroll
        for (int kf = 0; kf < KF; ++kf) {
            v2f a;
            a.x = ap[kf * 4 + kk];
            a.y = ap[kf * 4 + kk + 1];
            if (RELU) { a.x = fmaxf(a.x, 0.0f); a.y = fmaxf(a.y, 0.0f); }
            acc = __builtin_amdgcn_wmma_f32_16x16x4_f32(
                /*neg_a=*/false, a, /*neg_b=*/false, bfrag[kf],
                /*c_mod=*/(short)0, acc, /*reuse_a=*/false, /*reuse_b=*/false);
        }

        // C/D layout: VGPR i, lanes 0-15 -> M=i ; lanes 16-31 -> M=8+i ; N = lane&15
        const int rbase = r0 + ((lane >> 4) << 3);
        if (r0 + 16 <= M) {
            // hot path: full tile in bounds, straight-line stores
#pragma unroll
            for (int i = 0; i < 8; ++i) {
                const int r = rbase + i;
                const size_t idx = (size_t)r * FEAT + c0 + m;
                const float v = acc[i];
                out[idx] = v;
                const float dd = dis[r];
                initout[idx] = v * dd * dd + bcol;
            }
        } else {
            // cold path: partial last tile
#pragma unroll
            for (int i = 0; i < 8; ++i) {
                const int r = rbase + i;
                if (r < M) {
                    const size_t idx = (size_t)r * FEAT + c0 + m;
                    const float v = acc[i];
                    out[idx] = v;
                    const float dd = dis[r];
                    initout[idx] = v * dd * dd + bcol;
                }
            }
        }
    }
}

// ---------------- edge scatter: out[dst] += xw[src] * dis[src]*dis[dst] ----------------
// 16 lanes per edge, float4 per lane (b128 gather), f32 atomics resolve in L2.
__global__ void k_edge_scatter(const float4* __restrict__ xw4, const float* __restrict__ dis,
                               const long long* __restrict__ src, const long long* __restrict__ dst,
                               float* __restrict__ out, int E) {
    long long t = (long long)blockIdx.x * blockDim.x + threadIdx.x;
    int e   = (int)(t >> 4);
    int sub = (int)(t & 15);
    if (e >= E) return;
    int s = (int)src[e];
    int d = (int)dst[e];
    float nrm = dis[s] * dis[d];
    float4 v = xw4[(size_t)s * (FEAT / 4) + sub];
    float* q = out + (size_t)d * FEAT + sub * 4;
    atomicAdd(q + 0, v.x * nrm);
    atomicAdd(q + 1, v.y * nrm);
    atomicAdd(q + 2, v.z * nrm);
    atomicAdd(q + 3, v.w * nrm);
}

// ---------------- final projection: out = relu(h) @ Wo[64x2] + bo ----------------
__global__ void k_output(const float* __restrict__ h, const float* __restrict__ Wo,
                         const float* __restrict__ bo, float* __restrict__ out, int n) {
    int i = blockIdx.x * blockDim.x + threadIdx.x;
    if (i >= n) return;
    float a0 = bo[0], a1 = bo[1];
    const float4* hp4 = (const float4*)(h + (size_t)i * FEAT);
#pragma unroll
    for (int k4 = 0; k4 < FEAT / 4; ++k4) {
        float4 v = hp4[k4];
        float v0 = fmaxf(v.x, 0.0f), v1 = fmaxf(v.y, 0.0f);
        float v2 = fmaxf(v.z, 0.0f), v3 = fmaxf(v.w, 0.0f);
        int k = k4 * 4;
        a0 += v0 * Wo[2 * k + 0] + v1 * Wo[2 * k + 2] + v2 * Wo[2 * k + 4] + v3 * Wo[2 * k + 6];
        a1 += v0 * Wo[2 * k + 1] + v1 * Wo[2 * k + 3] + v2 * Wo[2 * k + 5] + v3 * Wo[2 * k + 7];
    }
    out[2 * i]     = a0;
    out[2 * i + 1] = a1;
}

extern "C" void kernel_launch(void* const* d_in, const int* in_sizes, int n_in,
                              void* d_out, int out_size, void* d_ws, size_t ws_size,
                              hipStream_t stream) {
    const float*     x   = (const float*)d_in[0];
    const long long* ei  = (const long long*)d_in[1];
    const float*     W1  = (const float*)d_in[2];
    const float*     b1  = (const float*)d_in[3];
    const float*     W2  = (const float*)d_in[4];
    const float*     b2  = (const float*)d_in[5];
    const float*     W3  = (const float*)d_in[6];
    const float*     b3  = (const float*)d_in[7];
    const float*     Wo  = (const float*)d_in[8];
    const float*     bo  = (const float*)d_in[9];

    const int N = in_sizes[0] / 32;   // 100000
    const int E = in_sizes[1] / 2;    // 1600000
    const long long* src = ei;
    const long long* dst = ei + E;

    // workspace: dis[N] | bufA[N*64] | bufB[N*64] | bufC[N*64]
    char* ws = (char*)d_ws;
    float* dis = (float*)ws;
    size_t off = (((size_t)N * 4) + 255) & ~(size_t)255;
    float* bufA = (float*)(ws + off); off += (size_t)N * FEAT * 4;
    float* bufB = (float*)(ws + off); off += (size_t)N * FEAT * 4;
    float* bufC = (float*)(ws + off);

    const int TB = 256;
    const int gN   = (N + TB - 1) / TB;
    const int gE   = (E + TB - 1) / TB;
    const int gE16 = (int)(((long long)E * 16 + TB - 1) / TB);
    const int gM   = (N + 63) / 64;   // 64 rows (4 tiles) per block

    // normalization: dis = rsqrt(1 + indegree)
    k_deg_init<<<gN, TB, 0, stream>>>(dis, N);
    k_deg_edges<<<gE, TB, 0, stream>>>(dst, dis, E);
    k_rsqrt_inplace<<<gN, TB, 0, stream>>>(dis, N);

    // layer 1: xw = x @ W1 (K=32) -> A, init B = xw*dis^2 + b1 ; scatter A -> B
    k_gemm_wmma_fused<32, false, 4><<<gM, 128, 0, stream>>>(x, W1, b1, dis, bufA, bufB, N);
    k_edge_scatter<<<gE16, TB, 0, stream>>>((const float4*)bufA, dis, src, dst, bufB, E);

    // layer 2: xw = relu(B) @ W2 (K=64) -> A, init C ; scatter A -> C
    k_gemm_wmma_fused<64, true, 4><<<gM, 128, 0, stream>>>(bufB, W2, b2, dis, bufA, bufC, N);
    k_edge_scatter<<<gE16, TB, 0, stream>>>((const float4*)bufA, dis, src, dst, bufC, E);

    // layer 3: xw = relu(C) @ W3 (K=64) -> A, init B ; scatter A -> B
    k_gemm_wmma_fused<64, true, 4><<<gM, 128, 0, stream>>>(bufC, W3, b3, dis, bufA, bufB, N);
    k_edge_scatter<<<gE16, TB, 0, stream>>>((const float4*)bufA, dis, src, dst, bufB, E);

    // output: relu(B) @ Wo + bo
    k_output<<<gN, TB, 0, stream>>>(bufB, Wo, bo, (float*)d_out, N);
}